// Featurizer_14645838479367
// MI455X (gfx1250) — compile-verified
//
#include <hip/hip_runtime.h>
#include <math.h>

// Problem constants (match reference)
#define Zb   4
#define Nn   512
#define Aa   14
#define AAa  21
#define Kk   30
#define Vv   256
#define EPSf 1e-6f

typedef __attribute__((ext_vector_type(2))) float v2f;
typedef __attribute__((ext_vector_type(8))) float v8f;

// ---------------------------------------------------------------------------
// Kernel 1: K-nearest-neighbor selection, replicating lax.top_k(-dists, K)
// semantics exactly (ascending distance, ties broken by ascending index).
// One thread per (z, i). CA positions + kp mask staged in LDS per block.
// Stores neighbor index j if edge is live, else -1.
// ---------------------------------------------------------------------------
__global__ __launch_bounds__(256) void knn_kernel(const float* __restrict__ C,
                                                  const unsigned char* __restrict__ kp,
                                                  int* __restrict__ nbr) {
    __shared__ float sx[Nn], sy[Nn], sz[Nn];
    __shared__ unsigned char skp[Nn];
    const int z   = blockIdx.x >> 1;          // 2 blocks per batch entry
    const int ib  = (blockIdx.x & 1) * 256;
    const int tid = threadIdx.x;

    for (int j = tid; j < Nn; j += 256) {
        const float* ca = C + ((z * Nn + j) * Aa + 1) * 3;  // atom 1 = CA
        sx[j] = ca[0]; sy[j] = ca[1]; sz[j] = ca[2];
        skp[j] = kp[z * Nn + j];
    }
    __syncthreads();

    const int i   = ib + tid;
    const float cx = sx[i], cy = sy[i], cz = sz[i];
    const bool kpi = (skp[i] != 0);

    float last_d = -INFINITY;
    int   last_j = -1;
    for (int k = 0; k < Kk; ++k) {
        float best_d = INFINITY;
        int   best_j = 0x7fffffff;
        for (int j = 0; j < Nn; ++j) {
            float dx = sx[j] - cx, dy = sy[j] - cy, dz = sz[j] - cz;
            float d  = sqrtf(dx * dx + dy * dy + dz * dz);
            if (d == 0.0f || kpi) d = INFINITY;   // reference masking
            // lexicographic key (d, j): strictly greater than previous pick,
            // minimal among candidates  ==  k-th element of sorted order
            bool gt = (d > last_d) || (d == last_d && j > last_j);
            bool lt = (d < best_d) || (d == best_d && j < best_j);
            if (gt && lt) { best_d = d; best_j = j; }
        }
        last_d = best_d; last_j = best_j;
        bool edge = !kpi && (skp[best_j] == 0);   // vals>0 always (0->inf mask)
        nbr[(z * Nn + i) * Kk + k] = edge ? best_j : -1;
    }
}

// ---------------------------------------------------------------------------
// Kernel 2: electrostatic voxel potential. One block per (z,n).
// Per neighbor, the 256x14 distance^2 matrix is 16 WMMA_F32_16X16X4_F32 tiles:
//   d2[v,a] = |p_v|^2 (C bias) + [-2p | 1] x [x ; |x|^2]   (A 16x4, B 4x16)
// All 30 neighbors' B operands + charges are pre-staged in LDS in exact
// per-lane WMMA layout, so the hot loop is barrier-free and branch-free:
//   ds_load_b64 (B pair) + ds_load_b32 (q) + 2x WMMA + 16x (rsq, min, fma).
// 1/max(sqrt(d2),eps) == min(rsqrt(d2), 1e6) for d2>=0  ->  one v_rsq_f32.
// Rare small-d2 elements are recomputed exactly for numerical safety.
// ---------------------------------------------------------------------------
__global__ __launch_bounds__(256) void potential_kernel(
    const float* __restrict__ C, const int* __restrict__ L,
    const unsigned char* __restrict__ amask,
    const float* __restrict__ amber, const float* __restrict__ voxel,
    const int* __restrict__ nbr, float* __restrict__ P) {

    __shared__ float s_pt[Vv * 3];     // voxel points
    __shared__ float s_pp[Vv];         // |p|^2
    __shared__ v2f   s_B[Kk][32];      // B operand, exact per-lane WMMA layout
    __shared__ float s_q[Kk][16];      // charges per neighbor per atom column
    __shared__ float s_out[Vv];

    const int zn  = blockIdx.x;
    const int z   = zn / Nn;
    const int tid = threadIdx.x;

    // ---- local frame (computed redundantly per thread; ~60 flops) ----
    const float* b = C + (size_t)zn * Aa * 3;
    float nx = b[0], ny = b[1], nz = b[2];
    float cax = b[3], cay = b[4], caz = b[5];
    float ccx = b[6], ccy = b[7], ccz = b[8];
    float b1x = cax - nx,  b1y = cay - ny,  b1z = caz - nz;
    float b2x = ccx - cax, b2y = ccy - cay, b2z = ccz - caz;
    float b3x = b1y * b2z - b1z * b2y;
    float b3y = b1z * b2x - b1x * b2z;
    float b3z = b1x * b2y - b1y * b2x;
    float cbx = cax - 0.58273431f * b2x + 0.56802827f * b1x - 0.54067466f * b3x;
    float cby = cay - 0.58273431f * b2y + 0.56802827f * b1y - 0.54067466f * b3y;
    float cbz = caz - 0.58273431f * b2z + 0.56802827f * b1z - 0.54067466f * b3z;
    float yux = cbx - cax, yuy = cby - cay, yuz = cbz - caz;
    float yin = 1.0f / fmaxf(sqrtf(yux * yux + yuy * yuy + yuz * yuz), EPSf);
    yux *= yin; yuy *= yin; yuz *= yin;
    float xrx = ccx - nx, xry = ccy - ny, xrz = ccz - nz;
    float pr  = xrx * yux + xry * yuy + xrz * yuz;
    float xvx = xrx - pr * yux, xvy = xry - pr * yuy, xvz = xrz - pr * yuz;
    float xin = 1.0f / fmaxf(sqrtf(xvx * xvx + xvy * xvy + xvz * xvz), EPSf);
    float xux = xvx * xin, xuy = xvy * xin, xuz = xvz * xin;
    float zux = xuy * yuz - xuz * yuy;
    float zuy = xuz * yux - xux * yuz;
    float zuz = xux * yuy - xuy * yux;

    // ---- this thread's voxel point ----
    float vx = voxel[tid * 3 + 0], vy = voxel[tid * 3 + 1], vz = voxel[tid * 3 + 2];
    float px = cbx + vx * xux + vy * yux + vz * zux;
    float py = cby + vx * xuy + vy * yuy + vz * zuy;
    float pz = cbz + vx * xuz + vy * yuz + vz * zuz;
    s_pt[tid * 3 + 0] = px; s_pt[tid * 3 + 1] = py; s_pt[tid * 3 + 2] = pz;
    s_pp[tid] = px * px + py * py + pz * pz;

    // ---- pre-stage ALL neighbors' B operands + charges (WMMA lane layout) --
    const int* nb = nbr + (size_t)zn * Kk;
    for (int s = tid; s < Kk * 32; s += 256) {
        int k = s >> 5, ln = s & 31, h = ln >> 4, a = ln & 15;
        int j = nb[k];
        float ax = 0.f, ay = 0.f, az = 0.f, q = 0.f;
        if (j >= 0 && a < Aa) {         // masked edge / pad column -> zeros
            const float* cj = C + ((size_t)(z * Nn + j) * Aa + a) * 3;
            ax = cj[0]; ay = cj[1]; az = cj[2];
            int Lj = L[z * Nn + j];
            int Lm = (Lj == -1) ? (AAa - 1) : Lj;
            q = amber[Lm * Aa + a] *
                (amask[(size_t)(z * Nn + j) * Aa + a] ? 1.0f : 0.0f);
        }
        // B 4x16: lanes 0-15 -> rows K=0,1 (x,y); lanes 16-31 -> K=2,3 (z,|x|^2)
        v2f bv;
        bv.x = h ? az : ax;
        bv.y = h ? (ax * ax + ay * ay + az * az) : ay;
        s_B[k][ln] = bv;
        if (h == 0) s_q[k][a] = q;
    }
    __syncthreads();

    const int wv = tid >> 5, lane = tid & 31, hf = lane >> 4, li = lane & 15;
    const int tbs[2] = { wv * 16, (wv + 8) * 16 };   // two voxel tiles per wave

    v2f aR[2]; v8f cR[2];
    float acc[2][8];
#pragma unroll
    for (int t = 0; t < 2; ++t) {
        int m = tbs[t] + li;
        float pxm = s_pt[m * 3 + 0], pym = s_pt[m * 3 + 1], pzm = s_pt[m * 3 + 2];
        // A 16x4 layout: lanes 0-15 -> K=0,1 ; lanes 16-31 -> K=2,3
        aR[t].x = hf ? (-2.0f * pzm) : (-2.0f * pxm);
        aR[t].y = hf ? 1.0f          : (-2.0f * pym);
#pragma unroll
        for (int r = 0; r < 8; ++r) {
            cR[t][r] = s_pp[tbs[t] + r + 8 * hf];    // C/D row = r + 8*half
            acc[t][r] = 0.0f;
        }
    }

    // ---- hot loop: barrier-free, branch-free (guard is dynamically rare) --
#pragma unroll 2
    for (int k = 0; k < Kk; ++k) {
        v2f  bR  = s_B[k][lane];
        float qli = s_q[k][li];
#pragma unroll
        for (int t = 0; t < 2; ++t) {
            v8f d = __builtin_amdgcn_wmma_f32_16x16x4_f32(
                false, aR[t], false, bR, (short)0, cR[t], false, false);
#pragma unroll
            for (int r = 0; r < 8; ++r) {
                float d2 = d[r];
                if (d2 < 1.0f) {         // exact recompute near cancellation
                    int m = tbs[t] + r + 8 * hf;
                    v2f   xy = s_B[k][li];        // (x, y) of atom li
                    float xz = s_B[k][16 + li].x; // z of atom li
                    float dx = s_pt[m * 3 + 0] - xy.x;
                    float dy = s_pt[m * 3 + 1] - xy.y;
                    float dz = s_pt[m * 3 + 2] - xz;
                    d2 = dx * dx + dy * dy + dz * dz;
                }
                // 1/max(sqrt(d2),eps) == min(rsqrt(d2), 1/eps) for d2 >= 0
                float inv = fminf(__builtin_amdgcn_rsqf(fmaxf(d2, 0.0f)),
                                  1.0f / EPSf);
                acc[t][r] += qli * inv;      // q=0 for pad/masked columns
            }
        }
    }

    // ---- reduce over the 16 atom columns (once, after all neighbors) ----
#pragma unroll
    for (int t = 0; t < 2; ++t) {
#pragma unroll
        for (int r = 0; r < 8; ++r) {
            float v = acc[t][r];
            v += __shfl_xor(v, 1, 32);
            v += __shfl_xor(v, 2, 32);
            v += __shfl_xor(v, 4, 32);
            v += __shfl_xor(v, 8, 32);   // stays inside each 16-lane half
            if (li == 0) s_out[tbs[t] + 8 * hf + r] = v;
        }
    }
    __syncthreads();
    P[(size_t)zn * Vv + tid] = s_out[tid];
}

// ---------------------------------------------------------------------------
extern "C" void kernel_launch(void* const* d_in, const int* in_sizes, int n_in,
                              void* d_out, int out_size, void* d_ws, size_t ws_size,
                              hipStream_t stream) {
    (void)in_sizes; (void)n_in; (void)out_size; (void)ws_size;
    const float*         C     = (const float*)d_in[0];
    const int*           L     = (const int*)d_in[1];
    const unsigned char* amask = (const unsigned char*)d_in[2];  // numpy bool
    const unsigned char* kp    = (const unsigned char*)d_in[3];  // numpy bool
    const float*         amber = (const float*)d_in[4];
    const float*         voxel = (const float*)d_in[5];
    float* P   = (float*)d_out;
    int*   nbr = (int*)d_ws;     // Z*N*K ints = 240 KB of workspace

    knn_kernel<<<Zb * 2, 256, 0, stream>>>(C, kp, nbr);
    potential_kernel<<<Zb * Nn, 256, 0, stream>>>(C, L, amask, amber, voxel, nbr, P);
}